// EnhancedMSTSN_14714557956228
// MI455X (gfx1250) — compile-verified
//
#include <hip/hip_runtime.h>
#include <math.h>

// Problem constants (match reference)
#define NN   4000
#define KK   20
#define BBATCH 2
#define TT   12
#define FIN_ 3
#define UU   32
#define EMB_ 16
#define H1_  4
#define EPS_ 1e-3f
#define INV_SQRT_U 0.1767766952966369f  // 1/sqrt(32)
#define NEG_BIG (-1.0e30f)

typedef float v2f __attribute__((ext_vector_type(2)));
typedef float v8f __attribute__((ext_vector_type(8)));

// ---------------- wave helpers (wave32) ----------------
__device__ __forceinline__ float wave_sum(float v){
#pragma unroll
  for(int m=16;m>0;m>>=1) v += __shfl_xor(v,m,32);
  return v;
}
__device__ __forceinline__ float half_max16(float v){ // reduce over 16-lane group
#pragma unroll
  for(int m=8;m>0;m>>=1) v = fmaxf(v,__shfl_xor(v,m,32));
  return v;
}
__device__ __forceinline__ float half_sum16(float v){
#pragma unroll
  for(int m=8;m>0;m>>=1) v += __shfl_xor(v,m,32);
  return v;
}
__device__ __forceinline__ float gelu_f(float x){
  return 0.5f*x*(1.0f+erff(x*0.70710678118654752f));
}

// ---------------- 1) normalize embeddings ----------------
__global__ void k_embn(const float* __restrict__ emb, float* __restrict__ embn){
  int n = blockIdx.x*blockDim.x + threadIdx.x;
  if(n>=NN) return;
  float v[EMB_]; float s=1e-12f;
#pragma unroll
  for(int e=0;e<EMB_;e++){ v[e]=emb[n*EMB_+e]; s=fmaf(v[e],v[e],s); }
  float r=rsqrtf(s);
#pragma unroll
  for(int e=0;e<EMB_;e++) embn[n*EMB_+e]=v[e]*r;
}

// ---------------- 2) fused sim-row + top-20 (avoids 64MB sim matrix) ----------------
__global__ void __launch_bounds__(256) k_topk(const float* __restrict__ embn, int* __restrict__ idx){
  __shared__ float row[NN];
  __shared__ float bval[256];
  __shared__ int   bidx[256];
  const int i = blockIdx.x, tid = threadIdx.x;
  float ei[EMB_];
#pragma unroll
  for(int e=0;e<EMB_;e++) ei[e]=embn[i*EMB_+e];
  for(int j=tid;j<NN;j+=256){
    float d=0.f;
#pragma unroll
    for(int e=0;e<EMB_;e++) d=fmaf(ei[e],embn[j*EMB_+e],d);
    row[j]=d;
  }
  __syncthreads();
  for(int it=0; it<KK; it++){
    float best=NEG_BIG; int bi=0x7fffffff;
    for(int j=tid;j<NN;j+=256){
      float v=row[j];
      if(v>best){best=v;bi=j;}   // j increases -> ties keep lowest index
    }
    bval[tid]=best; bidx[tid]=bi;
    __syncthreads();
    for(int s=128;s>0;s>>=1){
      if(tid<s){
        float v=bval[tid+s]; int j2=bidx[tid+s];
        if(v>bval[tid] || (v==bval[tid] && j2<bidx[tid])){ bval[tid]=v; bidx[tid]=j2; }
      }
      __syncthreads();
    }
    if(tid==0){ idx[i*KK+it]=bidx[0]; row[bidx[0]]=NEG_BIG; }
    __syncthreads();
  }
}

// ---------------- 3) GAT layer 1 (fully fused per node; FIN=3 so g recomputed) ----------------
__global__ void __launch_bounds__(32) k_gat1(
  const float* __restrict__ x, const int* __restrict__ idx,
  const float* __restrict__ W1, const float* __restrict__ asrc,
  const float* __restrict__ adst, const float* __restrict__ bias1,
  float* __restrict__ h1)
{
  const int lane=threadIdx.x, i=blockIdx.x;
  float Wl[FIN_][H1_], asl[H1_], adl[H1_];
#pragma unroll
  for(int f=0;f<FIN_;f++)
#pragma unroll
    for(int h=0;h<H1_;h++) Wl[f][h]=W1[(f*H1_+h)*UU+lane];
#pragma unroll
  for(int h=0;h<H1_;h++){ asl[h]=asrc[h*UU+lane]; adl[h]=adst[h*UU+lane]; }

  const float xi0=x[i*FIN_], xi1=x[i*FIN_+1], xi2=x[i*FIN_+2];
  float es[H1_];
#pragma unroll
  for(int h=0;h<H1_;h++){
    float gi = xi0*Wl[0][h]+xi1*Wl[1][h]+xi2*Wl[2][h];
    es[h]=wave_sum(gi*asl[h]);
  }
  __shared__ float eS[KK+1][H1_];
  __shared__ float alS[KK+1][H1_];
  for(int j=0;j<=KK;j++){
    int nb = (j<KK)? idx[i*KK+j] : i;
    float x0=x[nb*FIN_], x1=x[nb*FIN_+1], x2=x[nb*FIN_+2];
#pragma unroll
    for(int h=0;h<H1_;h++){
      float gn = x0*Wl[0][h]+x1*Wl[1][h]+x2*Wl[2][h];
      float ed = wave_sum(gn*adl[h]);
      float e = es[h]+ed;
      e = (e>0.f)? e : 0.2f*e;            // leaky_relu 0.2
      if(lane==0) eS[j][h]=e;
    }
  }
  __syncthreads();
  if(lane<H1_){
    const int h=lane;
    float m=NEG_BIG;
    for(int j=0;j<=KK;j++) m=fmaxf(m,eS[j][h]);
    float s=0.f;
    for(int j=0;j<=KK;j++){ float a=expf(eS[j][h]-m); alS[j][h]=a; s+=a; }
    float inv=1.f/s;
    for(int j=0;j<=KK;j++) alS[j][h]*=inv;
  }
  __syncthreads();
  float acc[H1_]={0.f,0.f,0.f,0.f};
  for(int j=0;j<=KK;j++){
    int nb=(j<KK)? idx[i*KK+j] : i;
    float x0=x[nb*FIN_], x1=x[nb*FIN_+1], x2=x[nb*FIN_+2];
#pragma unroll
    for(int h=0;h<H1_;h++){
      float gn=x0*Wl[0][h]+x1*Wl[1][h]+x2*Wl[2][h];
      acc[h]=fmaf(alS[j][h],gn,acc[h]);
    }
  }
#pragma unroll
  for(int h=0;h<H1_;h++){
    float o=acc[h]+bias1[h*UU+lane];
    h1[(size_t)i*(H1_*UU)+h*UU+lane]=fmaxf(o,0.f);   // relu
  }
}

// ---------------- generic fp32 WMMA GEMM: C[M,N] = A[M,K] @ B[K,N] (+bias) ----------------
// One wave computes one 16x16 tile via chained V_WMMA_F32_16X16X4_F32.
__global__ void __launch_bounds__(32) k_gemm_wmma(
  const float* __restrict__ A, int lda,
  const float* __restrict__ Bm, int ldb,
  float* __restrict__ C, int ldc,
  int Kdim, const float* __restrict__ bias)
{
  const int lane=threadIdx.x;
  const int r0=blockIdx.x*16, c0=blockIdx.y*16;
  const int mr=lane&15, kh=lane>>4;
  v8f acc = {0.f,0.f,0.f,0.f,0.f,0.f,0.f,0.f};
  for(int kk=0; kk<Kdim; kk+=4){
    v2f a, b;
    a.x = A[(size_t)(r0+mr)*lda + kk + kh*2];
    a.y = A[(size_t)(r0+mr)*lda + kk + kh*2 + 1];
    b.x = Bm[(size_t)(kk + kh*2    )*ldb + c0 + mr];
    b.y = Bm[(size_t)(kk + kh*2 + 1)*ldb + c0 + mr];
    acc = __builtin_amdgcn_wmma_f32_16x16x4_f32(false, a, false, b, (short)0, acc, false, false);
  }
  float bb = bias ? bias[c0+mr] : 0.f;
#pragma unroll
  for(int r=0;r<8;r++)
    C[(size_t)(r0 + kh*8 + r)*ldc + c0 + mr] = acc[r] + bb;
}

// ---------------- GAT2 attention logits ----------------
__global__ void k_e2(const float* __restrict__ g2, const float* __restrict__ a2s,
                     const float* __restrict__ a2d, float* __restrict__ e2s, float* __restrict__ e2d){
  int n = blockIdx.x*blockDim.x+threadIdx.x;
  if(n>=NN) return;
  float s1=0.f,s2=0.f;
  for(int k=0;k<UU;k++){ float g=g2[(size_t)n*UU+k]; s1=fmaf(g,a2s[k],s1); s2=fmaf(g,a2d[k],s2); }
  e2s[n]=s1; e2d[n]=s2;
}

// ---------------- GAT2 aggregation (single head) ----------------
__global__ void __launch_bounds__(256) k_gat2(
  const float* __restrict__ g2, const float* __restrict__ e2s, const float* __restrict__ e2d,
  const int* __restrict__ idx, const float* __restrict__ bias2,
  float* __restrict__ spOut)
{
  const int lane=threadIdx.x&31, wv=threadIdx.x>>5;
  const int i = blockIdx.x*8+wv;
  if(i>=NN) return;
  const float esi=e2s[i];
  float ev[KK+1]; float m=NEG_BIG;
#pragma unroll
  for(int j=0;j<=KK;j++){
    int nb=(j<KK)? idx[i*KK+j] : i;
    float e = esi + e2d[nb];
    e = (e>0.f)? e : 0.2f*e;
    ev[j]=e; m=fmaxf(m,e);
  }
  float s=0.f;
#pragma unroll
  for(int j=0;j<=KK;j++){ float a=expf(ev[j]-m); ev[j]=a; s+=a; }
  const float inv=1.f/s;
  float acc=0.f;
#pragma unroll
  for(int j=0;j<=KK;j++){
    int nb=(j<KK)? idx[i*KK+j] : i;
    acc=fmaf(ev[j]*inv, g2[(size_t)nb*UU+lane], acc);
  }
  spOut[(size_t)i*UU+lane]=acc+bias2[lane];
}

// ---------------- fused temporal transformer per (b,n) sequence (T=12, VALU) ----------------
__global__ void __launch_bounds__(32) k_temporal(
    const float* __restrict__ sp,
    const float* __restrict__ wq, const float* __restrict__ bq,
    const float* __restrict__ wk, const float* __restrict__ bk,
    const float* __restrict__ wv, const float* __restrict__ bv,
    const float* __restrict__ wo, const float* __restrict__ bo,
    const float* __restrict__ ln1g, const float* __restrict__ ln1b,
    const float* __restrict__ fw1, const float* __restrict__ fb1,
    const float* __restrict__ fw2, const float* __restrict__ fb2,
    const float* __restrict__ ln2g, const float* __restrict__ ln2b,
    float* __restrict__ sfeat, float* __restrict__ tfeat)
{
  __shared__ float tinS[TT][UU];
  __shared__ float qS[TT][2*UU];
  __shared__ float kS[TT][2*UU];
  __shared__ float vS[TT][2*UU];
  __shared__ float aS[2*TT][TT];
  __shared__ float oS[TT][2*UU];
  __shared__ float hS[TT][UU];
  __shared__ float hidS[TT][64];
  const int lane = threadIdx.x;
  const int bn = blockIdx.x;
  const int b = bn / NN, n = bn % NN;

  float tin[TT]; float sm=0.f;
#pragma unroll
  for(int t=0;t<TT;t++){
    float v = sp[((size_t)(b*TT+t)*NN + n)*UU + lane];
    tin[t]=v; tinS[t][lane]=v; sm+=v;
  }
  sfeat[(size_t)bn*UU + lane] = sm*(1.f/TT);
  __syncthreads();

  { // Q,K,V projections: lane owns head-dim k for both heads
    float a0[TT], a1[TT];
#pragma unroll
    for(int t=0;t<TT;t++){ a0[t]=bq[lane]; a1[t]=bq[32+lane]; }
    for(int d=0; d<UU; d++){
      float w0=wq[d*64+lane], w1=wq[d*64+32+lane];
#pragma unroll
      for(int t=0;t<TT;t++){ float xx=tinS[t][d]; a0[t]=fmaf(xx,w0,a0[t]); a1[t]=fmaf(xx,w1,a1[t]); }
    }
#pragma unroll
    for(int t=0;t<TT;t++){ qS[t][lane]=a0[t]; qS[t][32+lane]=a1[t]; }

#pragma unroll
    for(int t=0;t<TT;t++){ a0[t]=bk[lane]; a1[t]=bk[32+lane]; }
    for(int d=0; d<UU; d++){
      float w0=wk[d*64+lane], w1=wk[d*64+32+lane];
#pragma unroll
      for(int t=0;t<TT;t++){ float xx=tinS[t][d]; a0[t]=fmaf(xx,w0,a0[t]); a1[t]=fmaf(xx,w1,a1[t]); }
    }
#pragma unroll
    for(int t=0;t<TT;t++){ kS[t][lane]=a0[t]; kS[t][32+lane]=a1[t]; }

#pragma unroll
    for(int t=0;t<TT;t++){ a0[t]=bv[lane]; a1[t]=bv[32+lane]; }
    for(int d=0; d<UU; d++){
      float w0=wv[d*64+lane], w1=wv[d*64+32+lane];
#pragma unroll
      for(int t=0;t<TT;t++){ float xx=tinS[t][d]; a0[t]=fmaf(xx,w0,a0[t]); a1[t]=fmaf(xx,w1,a1[t]); }
    }
#pragma unroll
    for(int t=0;t<TT;t++){ vS[t][lane]=a0[t]; vS[t][32+lane]=a1[t]; }
  }
  __syncthreads();

  // scores + softmax: lanes 0..23 each own (h, t1)
  if(lane < 2*TT){
    const int h=lane/TT, t1=lane%TT;
    float sc[TT]; float mx=NEG_BIG;
#pragma unroll
    for(int t2=0;t2<TT;t2++){
      float d=0.f;
#pragma unroll
      for(int kd=0;kd<UU;kd++) d = fmaf(qS[t1][h*32+kd], kS[t2][h*32+kd], d);
      d *= INV_SQRT_U;
      sc[t2]=d; mx=fmaxf(mx,d);
    }
    float s=0.f;
#pragma unroll
    for(int t2=0;t2<TT;t2++){ float a=expf(sc[t2]-mx); sc[t2]=a; s+=a; }
    float inv=1.f/s;
#pragma unroll
    for(int t2=0;t2<TT;t2++) aS[lane][t2]=sc[t2]*inv;
  }
  __syncthreads();

  // o = a @ v
#pragma unroll
  for(int t=0;t<TT;t++){
    float o0=0.f,o1=0.f;
#pragma unroll
    for(int t2=0;t2<TT;t2++){
      o0 = fmaf(aS[t][t2],    vS[t2][lane],    o0);
      o1 = fmaf(aS[TT+t][t2], vS[t2][32+lane], o1);
    }
    oS[t][lane]=o0; oS[t][32+lane]=o1;
  }
  __syncthreads();

  // out projection + residual + LN1
  float hv[TT];
  {
    float acc[TT];
#pragma unroll
    for(int t=0;t<TT;t++) acc[t]=bo[lane];
    for(int hk=0;hk<64;hk++){
      float w=wo[hk*32+lane];
#pragma unroll
      for(int t=0;t<TT;t++) acc[t]=fmaf(oS[t][hk],w,acc[t]);
    }
#pragma unroll
    for(int t=0;t<TT;t++){
      float y = tin[t]+acc[t];
      float mu = wave_sum(y)*(1.f/32.f);
      float d = y-mu;
      float var = wave_sum(d*d)*(1.f/32.f);
      float nv = d*rsqrtf(var+EPS_)*ln1g[lane]+ln1b[lane];
      hv[t]=nv; hS[t][lane]=nv;
    }
  }
  __syncthreads();

  // FFN: 32 -> 64 (gelu) -> 32
  {
    float a0[TT],a1[TT];
#pragma unroll
    for(int t=0;t<TT;t++){ a0[t]=fb1[lane]; a1[t]=fb1[32+lane]; }
    for(int d=0;d<UU;d++){
      float w0=fw1[d*64+lane], w1=fw1[d*64+32+lane];
#pragma unroll
      for(int t=0;t<TT;t++){ float xx=hS[t][d]; a0[t]=fmaf(xx,w0,a0[t]); a1[t]=fmaf(xx,w1,a1[t]); }
    }
#pragma unroll
    for(int t=0;t<TT;t++){ hidS[t][lane]=gelu_f(a0[t]); hidS[t][32+lane]=gelu_f(a1[t]); }
  }
  __syncthreads();
  {
    float acc[TT];
#pragma unroll
    for(int t=0;t<TT;t++) acc[t]=fb2[lane];
    for(int c=0;c<64;c++){
      float w=fw2[c*32+lane];
#pragma unroll
      for(int t=0;t<TT;t++) acc[t]=fmaf(hidS[t][c],w,acc[t]);
    }
    float tm=0.f;
#pragma unroll
    for(int t=0;t<TT;t++){
      float y = hv[t]+acc[t];
      float mu = wave_sum(y)*(1.f/32.f);
      float d=y-mu;
      float var = wave_sum(d*d)*(1.f/32.f);
      tm += d*rsqrtf(var+EPS_)*ln2g[lane]+ln2b[lane];
    }
    tfeat[(size_t)bn*UU+lane]=tm*(1.f/TT);
  }
}

// ---------------- flash cross-attention (s=t=4000) with fp32 WMMA + online softmax ----------------
__global__ void __launch_bounds__(32) k_flash(
  const float* __restrict__ cq, const float* __restrict__ ck, const float* __restrict__ cv,
  float* __restrict__ co)
{
  const int lane=threadIdx.x;
  const int st=blockIdx.x, h=blockIdx.y, b=blockIdx.z;
  const int mr=lane&15, kh=lane>>4;
  const float* Q  = cq + ((size_t)(b*NN + st*16))*64 + h*32;
  const float* Kb = ck + (size_t)b*NN*64 + h*32;
  const float* Vb = cv + (size_t)b*NN*64 + h*32;
  v8f o0={0.f,0.f,0.f,0.f,0.f,0.f,0.f,0.f};
  v8f o1={0.f,0.f,0.f,0.f,0.f,0.f,0.f,0.f};
  float mrun[8], lrun[8];
#pragma unroll
  for(int r=0;r<8;r++){ mrun[r]=NEG_BIG; lrun[r]=0.f; }
  __shared__ float pS[16][17];

  // preload Q A-fragments (K=32 -> 8 frags), reused for every key tile
  v2f aq[8];
#pragma unroll
  for(int kk=0;kk<8;kk++){
    aq[kk].x = Q[(size_t)mr*64 + kk*4 + kh*2];
    aq[kk].y = Q[(size_t)mr*64 + kk*4 + kh*2 + 1];
  }

  for(int tt=0; tt<NN/16; tt++){
    const float* Kt = Kb + (size_t)tt*16*64;
    const float* Vt = Vb + (size_t)tt*16*64;
    // S = Q @ K^T  (16x16)
    v8f s={0.f,0.f,0.f,0.f,0.f,0.f,0.f,0.f};
#pragma unroll
    for(int kk=0;kk<8;kk++){
      v2f bf;
      bf.x = Kt[(size_t)mr*64 + kk*4 + kh*2];
      bf.y = Kt[(size_t)mr*64 + kk*4 + kh*2 + 1];
      s = __builtin_amdgcn_wmma_f32_16x16x4_f32(false, aq[kk], false, bf, (short)0, s, false, false);
    }
    // online softmax on D-layout rows (row = kh*8+r, cols across 16-lane group)
#pragma unroll
    for(int r=0;r<8;r++){
      float sv = s[r]*INV_SQRT_U;
      float mx = half_max16(sv);
      float mnew = fmaxf(mrun[r], mx);
      float corr = expf(mrun[r]-mnew);
      float p = expf(sv-mnew);
      float rs = half_sum16(p);
      lrun[r]=lrun[r]*corr+rs;
      mrun[r]=mnew;
      o0[r]*=corr; o1[r]*=corr;
      pS[kh*8+r][mr]=p;                 // re-fragment P through LDS
    }
    __syncthreads();
    // O += P @ V  (K=16, two 16-col output tiles)
#pragma unroll
    for(int kk=0;kk<4;kk++){
      v2f ap;
      ap.x = pS[mr][kk*4+kh*2];
      ap.y = pS[mr][kk*4+kh*2+1];
      v2f bv0, bv1;
      bv0.x = Vt[(size_t)(kk*4+kh*2  )*64 + mr];
      bv0.y = Vt[(size_t)(kk*4+kh*2+1)*64 + mr];
      bv1.x = Vt[(size_t)(kk*4+kh*2  )*64 + 16 + mr];
      bv1.y = Vt[(size_t)(kk*4+kh*2+1)*64 + 16 + mr];
      o0 = __builtin_amdgcn_wmma_f32_16x16x4_f32(false, ap, false, bv0, (short)0, o0, false, false);
      o1 = __builtin_amdgcn_wmma_f32_16x16x4_f32(false, ap, false, bv1, (short)0, o1, false, false);
    }
    __syncthreads();
  }
  float* Co = co + ((size_t)(b*NN + st*16))*64 + h*32;
#pragma unroll
  for(int r=0;r<8;r++){
    float inv=1.f/lrun[r];
    Co[(size_t)(kh*8+r)*64 + mr]      = o0[r]*inv;
    Co[(size_t)(kh*8+r)*64 + 16 + mr] = o1[r]*inv;
  }
}

// ---------------- epilogue: output projection + LN + dense ----------------
__global__ void __launch_bounds__(32) k_final(
  const float* __restrict__ co, const float* __restrict__ wo, const float* __restrict__ bo,
  const float* __restrict__ lng, const float* __restrict__ lnb,
  const float* __restrict__ fdw, const float* __restrict__ fdb,
  float* __restrict__ out)
{
  const int lane=threadIdx.x, bn=blockIdx.x;
  float acc=bo[lane];
  for(int hk=0;hk<64;hk++) acc=fmaf(co[(size_t)bn*64+hk], wo[hk*32+lane], acc);
  float mu=wave_sum(acc)*(1.f/32.f);
  float d=acc-mu;
  float var=wave_sum(d*d)*(1.f/32.f);
  float f=d*rsqrtf(var+EPS_)*lng[lane]+lnb[lane];
  float y=wave_sum(f*fdw[lane]);
  if(lane==0) out[bn]=y+fdb[0];
}

// ---------------- host orchestration ----------------
extern "C" void kernel_launch(void* const* d_in, const int* in_sizes, int n_in,
                              void* d_out, int out_size, void* d_ws, size_t ws_size,
                              hipStream_t stream) {
  (void)in_sizes; (void)n_in; (void)out_size; (void)ws_size;
  const float* x    = (const float*)d_in[0];
  const float* emb  = (const float*)d_in[1];
  const float* g1W  = (const float*)d_in[2];
  const float* g1as = (const float*)d_in[3];
  const float* g1ad = (const float*)d_in[4];
  const float* g1b  = (const float*)d_in[5];
  const float* g2W  = (const float*)d_in[6];
  const float* g2as = (const float*)d_in[7];
  const float* g2ad = (const float*)d_in[8];
  const float* g2b  = (const float*)d_in[9];
  const float* t_wq = (const float*)d_in[10];
  const float* t_bq = (const float*)d_in[11];
  const float* t_wk = (const float*)d_in[12];
  const float* t_bk = (const float*)d_in[13];
  const float* t_wv = (const float*)d_in[14];
  const float* t_bv = (const float*)d_in[15];
  const float* t_wo = (const float*)d_in[16];
  const float* t_bo = (const float*)d_in[17];
  const float* ln1g = (const float*)d_in[18];
  const float* ln1b = (const float*)d_in[19];
  const float* fw1  = (const float*)d_in[20];
  const float* fb1  = (const float*)d_in[21];
  const float* fw2  = (const float*)d_in[22];
  const float* fb2  = (const float*)d_in[23];
  const float* ln2g = (const float*)d_in[24];
  const float* ln2b = (const float*)d_in[25];
  const float* c_wq = (const float*)d_in[26];
  const float* c_bq = (const float*)d_in[27];
  const float* c_wk = (const float*)d_in[28];
  const float* c_bk = (const float*)d_in[29];
  const float* c_wv = (const float*)d_in[30];
  const float* c_bv = (const float*)d_in[31];
  const float* c_wo = (const float*)d_in[32];
  const float* c_bo = (const float*)d_in[33];
  const float* ln_g = (const float*)d_in[34];
  const float* ln_b = (const float*)d_in[35];
  const float* fd_w = (const float*)d_in[36];
  const float* fd_b = (const float*)d_in[37];
  float* out = (float*)d_out;

  // workspace carve-up (~25 MB total)
  float* ws = (float*)d_ws;
  size_t off = 0;
  float* embn = ws+off;          off += (size_t)NN*EMB_;
  int*   idx  = (int*)(ws+off);  off += (size_t)NN*KK;
  float* h1   = ws+off;          off += (size_t)NN*(H1_*UU);
  float* g2   = ws+off;          off += (size_t)NN*UU;
  float* e2s  = ws+off;          off += NN;
  float* e2d  = ws+off;          off += NN;
  float* sp   = ws+off;          off += (size_t)BBATCH*TT*NN*UU;
  float* sfeat= ws+off;          off += (size_t)BBATCH*NN*UU;
  float* tfeat= ws+off;          off += (size_t)BBATCH*NN*UU;
  float* cq   = ws+off;          off += (size_t)BBATCH*NN*64;
  float* ckb  = ws+off;          off += (size_t)BBATCH*NN*64;
  float* cvb  = ws+off;          off += (size_t)BBATCH*NN*64;
  float* co   = ws+off;          off += (size_t)BBATCH*NN*64;

  // adjacency
  k_embn<<<(NN+255)/256, 256, 0, stream>>>(emb, embn);
  k_topk<<<NN, 256, 0, stream>>>(embn, idx);

  // spatial GATs per (b,t) graph; scratch reused across stream-ordered launches
  for(int bt=0; bt<BBATCH*TT; bt++){
    k_gat1<<<NN, 32, 0, stream>>>(x + (size_t)bt*NN*FIN_, idx, g1W, g1as, g1ad, g1b, h1);
    k_gemm_wmma<<<dim3(NN/16, UU/16), 32, 0, stream>>>(h1, H1_*UU, g2W, UU, g2, UU, H1_*UU, nullptr);
    k_e2<<<(NN+255)/256, 256, 0, stream>>>(g2, g2as, g2ad, e2s, e2d);
    k_gat2<<<(NN+7)/8, 256, 0, stream>>>(g2, e2s, e2d, idx, g2b, sp + (size_t)bt*NN*UU);
  }

  // fused temporal transformer; also produces sfeat (mean over T of sp)
  k_temporal<<<BBATCH*NN, 32, 0, stream>>>(sp,
      t_wq,t_bq, t_wk,t_bk, t_wv,t_bv, t_wo,t_bo,
      ln1g,ln1b, fw1,fb1, fw2,fb2, ln2g,ln2b,
      sfeat, tfeat);

  // cross-attention projections (WMMA GEMMs) and flash attention
  k_gemm_wmma<<<dim3(BBATCH*NN/16, 64/16), 32, 0, stream>>>(sfeat, UU, c_wq, 64, cq,  64, UU, c_bq);
  k_gemm_wmma<<<dim3(BBATCH*NN/16, 64/16), 32, 0, stream>>>(tfeat, UU, c_wk, 64, ckb, 64, UU, c_bk);
  k_gemm_wmma<<<dim3(BBATCH*NN/16, 64/16), 32, 0, stream>>>(tfeat, UU, c_wv, 64, cvb, 64, UU, c_bv);
  k_flash<<<dim3(NN/16, 2, BBATCH), 32, 0, stream>>>(cq, ckb, cvb, co);

  // output projection + LN + dense
  k_final<<<BBATCH*NN, 32, 0, stream>>>(co, c_wo, c_bo, ln_g, ln_b, fd_w, fd_b, out);
}